// SingleQAttentionSE3_53498112639496
// MI455X (gfx1250) — compile-verified
//
#include <hip/hip_runtime.h>

typedef float v2f __attribute__((ext_vector_type(2)));
typedef float v8f __attribute__((ext_vector_type(8)));

#define BATCH 8
#define T_PTS 8192
#define NF0   128
#define NF1   128
#define NH    8
#define NDH   64
#define X1W   (3 * NF1)        // 384 columns of x1 per point
#define SCOLS (NF0 + X1W)      // 512 combined value columns

// workspace layout (float offsets)
#define WQP_OFF 0                          // padded wq: 16 x 128  (h >= 8 rows are zero)
#define S_OFF   2048                       // s: B x 8 x 512 partial sums
#define LOG_OFF (2048 + 32768)             // logits/attn padded: B x 16 x T

// ---------------- kernel 0: wq = W_k @ q (padded to 16 heads), zero s ----------------
__global__ void k0_prep(const float* __restrict__ Wk, const float* __restrict__ q,
                        float* __restrict__ ws) {
    int idx = blockIdx.x * blockDim.x + threadIdx.x;
    if (idx < 16 * NF0) {
        int h = idx >> 7, d = idx & 127;
        float acc = 0.f;
        if (h < NH) {
            const float* wr = Wk + (size_t)(h * NF0 + d) * NDH;
            const float* qr = q + h * NDH;
            #pragma unroll 8
            for (int e = 0; e < NDH; ++e) acc += wr[e] * qr[e];
        }
        ws[WQP_OFF + idx] = acc;
    } else if (idx < 16 * NF0 + BATCH * NH * SCOLS) {
        ws[S_OFF + (idx - 16 * NF0)] = 0.f;
    }
}

// ---------------- kernel 1: logits[b,h,t] = 0.125 * x0[b,t,:] . wq[h,:]  (WMMA) ----------------
// grid (64, B), 256 threads (8 waves); wave handles one 16-t tile, K=128 in 32 steps of 4.
__global__ void k1_logits(const float* __restrict__ x0, float* __restrict__ ws) {
    const float* wqp = ws + WQP_OFF;
    float* logitsP   = ws + LOG_OFF;
    int b    = blockIdx.y;
    int wave = threadIdx.x >> 5;
    int lane = threadIdx.x & 31;
    int lm   = lane & 15;
    int lg   = lane >> 4;                 // 0: K=0,1  1: K=2,3
    int tbase = (blockIdx.x * 8 + wave) * 16;

    // A tile (16 t x 4 d): lane holds row t=tbase+lm, elems k+2*lg, k+2*lg+1
    const float* xrow = x0 + ((size_t)(b * T_PTS + tbase + lm)) * NF0 + 2 * lg;
    // B tile (4 d x 16 h): lane holds col h=lm, elems k+2*lg, k+2*lg+1
    const float* wrow = wqp + lm * NF0 + 2 * lg;

    v8f c = {0.f, 0.f, 0.f, 0.f, 0.f, 0.f, 0.f, 0.f};
    #pragma unroll 4
    for (int k = 0; k < NF0; k += 4) {
        v2f a  = *(const v2f*)(xrow + k);
        v2f bm = *(const v2f*)(wrow + k);
        c = __builtin_amdgcn_wmma_f32_16x16x4_f32(false, a, false, bm, (short)0, c,
                                                  false, false);
    }
    // C: vgpr r -> t = tbase + r + 8*lg (lanes 16-31 hold M=r+8); N = lm = h
    float* out = logitsP + ((size_t)(b * 16 + lm)) * T_PTS + tbase + lg * 8;
    #pragma unroll
    for (int r = 0; r < 8; ++r) out[r] = c[r] * 0.125f;
}

// ---------------- kernel 2: row softmax over T per (b,h); zero padded h rows ----------------
__global__ void k2_softmax(float* __restrict__ ws) {
    float* row = ws + LOG_OFF + (size_t)blockIdx.x * T_PTS;
    int h   = blockIdx.x & 15;
    int tid = threadIdx.x;
    __shared__ float red[256];
    if (h >= NH) {
        for (int i = tid; i < T_PTS; i += 256) row[i] = 0.f;
        return;
    }
    float m = -3.402823466e38f;
    for (int i = tid; i < T_PTS; i += 256) m = fmaxf(m, row[i]);
    red[tid] = m; __syncthreads();
    for (int off = 128; off > 0; off >>= 1) {
        if (tid < off) red[tid] = fmaxf(red[tid], red[tid + off]);
        __syncthreads();
    }
    float M = red[0]; __syncthreads();
    float sum = 0.f;
    for (int i = tid; i < T_PTS; i += 256) {
        float e = __expf(row[i] - M);
        row[i] = e; sum += e;
    }
    red[tid] = sum; __syncthreads();
    for (int off = 128; off > 0; off >>= 1) {
        if (tid < off) red[tid] += red[tid + off];
        __syncthreads();
    }
    float inv = 1.f / red[0];
    for (int i = tid; i < T_PTS; i += 256) row[i] *= inv;
}

// ---------------- kernel 3: s[b,h,n] += sum_t attn[b,h,t] * X[b,t,n]  (WMMA) ----------------
// grid (32, B), 256 threads. Block = one 256-t chunk; wave w covers cols [64w, 64w+64) (4 tiles).
__global__ void k3_accum(const float* __restrict__ x0, const float* __restrict__ x1,
                         float* __restrict__ ws) {
    const float* attnP = ws + LOG_OFF;
    float* s = ws + S_OFF;
    int b      = blockIdx.y;
    int tstart = blockIdx.x * 256;
    int wave   = threadIdx.x >> 5;
    int lane   = threadIdx.x & 31;
    int lm     = lane & 15;
    int lg     = lane >> 4;

    // A tile (16 h x 4 t): lane holds row h=lm; elems t=kt+2*lg, kt+2*lg+1 (contiguous)
    const float* arow = attnP + ((size_t)(b * 16 + lm)) * T_PTS + 2 * lg;

    const float* bbase[4];
    int bstride[4], ncol[4];
    #pragma unroll
    for (int tt = 0; tt < 4; ++tt) {
        int n = wave * 64 + tt * 16 + lm;   // column for this lane; per-tile x0/x1 choice is wave-uniform
        ncol[tt] = n;
        if (n < NF0) { bbase[tt] = x0 + (size_t)b * T_PTS * NF0 + n;          bstride[tt] = NF0; }
        else         { bbase[tt] = x1 + (size_t)b * T_PTS * X1W + (n - NF0);  bstride[tt] = X1W; }
    }

    v8f c0 = {0.f,0.f,0.f,0.f,0.f,0.f,0.f,0.f};
    v8f c1 = c0, c2 = c0, c3 = c0;

    for (int kt = tstart; kt < tstart + 256; kt += 4) {
        v2f a = *(const v2f*)(arow + kt);
        int row = kt + 2 * lg;
        {   const float* p = bbase[0] + (size_t)row * bstride[0];
            v2f bm; bm.x = p[0]; bm.y = p[bstride[0]];
            c0 = __builtin_amdgcn_wmma_f32_16x16x4_f32(false, a, false, bm, (short)0, c0, false, false); }
        {   const float* p = bbase[1] + (size_t)row * bstride[1];
            v2f bm; bm.x = p[0]; bm.y = p[bstride[1]];
            c1 = __builtin_amdgcn_wmma_f32_16x16x4_f32(false, a, false, bm, (short)0, c1, false, false); }
        {   const float* p = bbase[2] + (size_t)row * bstride[2];
            v2f bm; bm.x = p[0]; bm.y = p[bstride[2]];
            c2 = __builtin_amdgcn_wmma_f32_16x16x4_f32(false, a, false, bm, (short)0, c2, false, false); }
        {   const float* p = bbase[3] + (size_t)row * bstride[3];
            v2f bm; bm.x = p[0]; bm.y = p[bstride[3]];
            c3 = __builtin_amdgcn_wmma_f32_16x16x4_f32(false, a, false, bm, (short)0, c3, false, false); }
    }

    // C: vgpr r -> h = r (lanes 0-15) / r+8 (lanes 16-31, padded -> discard); N = col
    if (lane < 16) {
        #pragma unroll
        for (int r = 0; r < 8; ++r) {
            float* srow = s + ((size_t)(b * NH + r)) * SCOLS;
            atomicAdd(srow + ncol[0], c0[r]);
            atomicAdd(srow + ncol[1], c1[r]);
            atomicAdd(srow + ncol[2], c2[r]);
            atomicAdd(srow + ncol[3], c3[r]);
        }
    }
}

// ---------------- kernel 4: o = Wv @ s, out = Wo @ o  (tiny epilogue, one block per b) ----------------
__global__ void k4_final(const float* __restrict__ ws, const float* __restrict__ Wv0,
                         const float* __restrict__ Wv1, const float* __restrict__ Wo0,
                         const float* __restrict__ Wo1, float* __restrict__ out) {
    int b = blockIdx.x, tid = threadIdx.x;
    const float* s = ws + S_OFF;
    __shared__ float o[2048];              // o0: [0,512) by he; o1: 512 + he*3 + i

    for (int v = tid; v < 512; v += 256) {
        int h = v >> 6, e = v & 63;
        const float* wv0 = Wv0 + (size_t)(h * NDH + e) * NF0;
        const float* wv1 = Wv1 + (size_t)(h * NDH + e) * NF1;
        const float* sp  = s + (size_t)(b * NH + h) * SCOLS;
        float a0 = 0.f, ax = 0.f, ay = 0.f, az = 0.f;
        for (int d = 0; d < NF0; ++d) {
            a0 += wv0[d] * sp[d];
            float w1 = wv1[d];
            const float* s1 = sp + NF0 + d * 3;
            ax += w1 * s1[0]; ay += w1 * s1[1]; az += w1 * s1[2];
        }
        o[v] = a0;
        o[512 + v * 3 + 0] = ax;
        o[512 + v * 3 + 1] = ay;
        o[512 + v * 3 + 2] = az;
    }
    __syncthreads();

    for (int v = tid; v < 512; v += 256) {
        if (v < 128) {
            int f = v;
            const float* w = Wo0 + (size_t)f * (NH * NDH);
            float acc = 0.f;
            for (int he = 0; he < NH * NDH; ++he) acc += w[he] * o[he];
            out[b * NF0 + f] = acc;
        } else {
            int u = v - 128, f = u / 3, i = u % 3;
            const float* w = Wo1 + (size_t)f * (NH * NDH);
            float acc = 0.f;
            for (int he = 0; he < NH * NDH; ++he) acc += w[he] * o[512 + he * 3 + i];
            out[BATCH * NF0 + b * (NF1 * 3) + f * 3 + i] = acc;
        }
    }
}

extern "C" void kernel_launch(void* const* d_in, const int* in_sizes, int n_in,
                              void* d_out, int out_size, void* d_ws, size_t ws_size,
                              hipStream_t stream) {
    (void)in_sizes; (void)n_in; (void)out_size; (void)ws_size;
    const float* x0  = (const float*)d_in[0];
    const float* x1  = (const float*)d_in[1];
    const float* q   = (const float*)d_in[2];
    const float* Wk  = (const float*)d_in[3];
    const float* Wv0 = (const float*)d_in[4];
    const float* Wv1 = (const float*)d_in[5];
    const float* Wo0 = (const float*)d_in[6];
    const float* Wo1 = (const float*)d_in[7];
    float* out = (float*)d_out;
    float* ws  = (float*)d_ws;

    k0_prep   <<<136, 256, 0, stream>>>(Wk, q, ws);
    k1_logits <<<dim3(64, BATCH), 256, 0, stream>>>(x0, ws);
    k2_softmax<<<BATCH * 16, 256, 0, stream>>>(ws);
    k3_accum  <<<dim3(32, BATCH), 256, 0, stream>>>(x0, x1, ws);
    k4_final  <<<BATCH, 256, 0, stream>>>(ws, Wv0, Wv1, Wo0, Wo1, out);
}